// SimpleGCN_40484361732199
// MI455X (gfx1250) — compile-verified
//
#include <hip/hip_runtime.h>
#include <hip/hip_bf16.h>

typedef float v2f __attribute__((ext_vector_type(2)));
typedef float v4f __attribute__((ext_vector_type(4)));
typedef float v8f __attribute__((ext_vector_type(8)));

#define N_NODES 100000
#define N_EDGES 1600000
#define K_DIM   128   // inner dimension for both layers

// ---------------------------------------------------------------------------
// Degree / normalization kernels
// ---------------------------------------------------------------------------
__global__ void gcn_deg_count(const int* __restrict__ dst, float* __restrict__ deg, int e) {
    int i = blockIdx.x * blockDim.x + threadIdx.x;
    if (i < e) atomicAdd(&deg[dst[i]], 1.0f);
}

__global__ void gcn_deg_to_dinv(float* __restrict__ d, int n) {
    int i = blockIdx.x * blockDim.x + threadIdx.x;
    if (i < n) d[i] = rsqrtf(d[i] + 1.0f);   // +1 self-loop; always > 0
}

// ---------------------------------------------------------------------------
// fp32 GEMM via V_WMMA_F32_16X16X4_F32.
// X: [nrows x 128] row-major, W: [128 x COUT] row-major, Y: [nrows x COUT].
// One wave per 16-row M tile; all COUT/16 N-tile accumulators in registers;
// A fragment reused across N tiles.  W staged in LDS in K-pair-interleaved
// layout so each B fragment is one aligned ds_load_b64 (no repacking movs):
//   sW2[kp*COUT + col] = { W[2kp][col], W[2kp+1][col] }
// ---------------------------------------------------------------------------
template <int COUT>
__global__ __launch_bounds__(256) void gcn_gemm_wmma(const float* __restrict__ X,
                                                     const float* __restrict__ W,
                                                     float* __restrict__ Y,
                                                     int nrows) {
    __shared__ v2f sW2[(K_DIM / 2) * COUT];

    const int tid = threadIdx.x;
    // Cooperative staged copy of W into LDS, interleaving K pairs.
    for (int i = tid; i < (K_DIM / 2) * COUT; i += 256) {
        const int kp  = i / COUT;
        const int col = i - kp * COUT;
        v2f p;
        p.x = W[(2 * kp + 0) * COUT + col];
        p.y = W[(2 * kp + 1) * COUT + col];
        sW2[i] = p;
    }
    __syncthreads();

    const int wave = tid >> 5;
    const int lane = tid & 31;
    const int tile = blockIdx.x * 8 + wave;      // global 16-row tile index
    if (tile * 16 >= nrows) return;              // whole-wave exit (EXEC stays all-ones)

    const int row0 = tile * 16;
    const int m    = lane & 15;                  // row (A) / col (B) within tile
    const int kh   = lane >> 4;                  // 0: K={k,k+1}, 1: K={k+2,k+3}
    const float* xrow = X + (size_t)(row0 + m) * K_DIM;

    v8f acc[COUT / 16] = {};

    for (int ks = 0; ks < K_DIM / 4; ++ks) {
        const int k0 = ks * 4 + kh * 2;          // even -> 8B aligned
        // A fragment: lane holds X[row0+m][k0], X[row0+m][k0+1]
        v2f a = *(const v2f*)(xrow + k0);
        const v2f* brow = sW2 + (size_t)(ks * 2 + kh) * COUT + m;
#pragma unroll
        for (int t = 0; t < COUT / 16; ++t) {
            v2f b = brow[t * 16];                // {W[k0][col], W[k0+1][col]}
            acc[t] = __builtin_amdgcn_wmma_f32_16x16x4_f32(
                false, a, false, b, (short)0, acc[t], false, false);
        }
    }

    // C/D layout: lanes 0-15 hold rows 0-7 (VGPR r -> row r), lanes 16-31 rows 8-15.
    const int rbase = row0 + (kh << 3);
    const int ncol  = lane & 15;
#pragma unroll
    for (int t = 0; t < COUT / 16; ++t) {
#pragma unroll
        for (int r = 0; r < 8; ++r) {
            Y[(size_t)(rbase + r) * COUT + t * 16 + ncol] = acc[t][r];
        }
    }
}

// ---------------------------------------------------------------------------
// Edge scatter: agg[dst] += h[src] * dinv[src]*dinv[dst].  One wave per edge,
// each lane handles C/32 contiguous features (coalesced loads + atomics).
// ---------------------------------------------------------------------------
template <int C>
__global__ __launch_bounds__(256) void gcn_edge_scatter(const int* __restrict__ src,
                                                        const int* __restrict__ dst,
                                                        const float* __restrict__ dinv,
                                                        const float* __restrict__ h,
                                                        float* __restrict__ agg,
                                                        int e) {
    const int edge = blockIdx.x * 8 + (threadIdx.x >> 5);
    if (edge >= e) return;
    const int lane = threadIdx.x & 31;
    const int s = src[edge];
    const int d = dst[edge];
    const float norm = dinv[s] * dinv[d];

    constexpr int PER = C / 32;  // 4 (C=128) or 2 (C=64)
    const float* hp = h + (size_t)s * C + lane * PER;
    float*       ap = agg + (size_t)d * C + lane * PER;

    float v[PER];
#pragma unroll
    for (int j = 0; j < PER; ++j) v[j] = hp[j];
#pragma unroll
    for (int j = 0; j < PER; ++j) atomicAdd(&ap[j], v[j] * norm);
}

// ---------------------------------------------------------------------------
// Finalize: out = act(agg + h*dinv^2 + bias).  out may alias agg (elementwise).
// ---------------------------------------------------------------------------
template <int C, bool RELU>
__global__ void gcn_finalize(const float* agg, const float* __restrict__ h,
                             const float* __restrict__ dinv,
                             const float* __restrict__ bias,
                             float* out, int n) {
    size_t idx = (size_t)blockIdx.x * blockDim.x + threadIdx.x;
    if (idx >= (size_t)n * C) return;
    const int node = (int)(idx / C);
    const int f    = (int)(idx % C);
    const float di = dinv[node];
    float v = agg[idx] + h[idx] * di * di + bias[f];
    if (RELU) v = fmaxf(v, 0.0f);
    out[idx] = v;
}

// ---------------------------------------------------------------------------
extern "C" void kernel_launch(void* const* d_in, const int* in_sizes, int n_in,
                              void* d_out, int out_size, void* d_ws, size_t ws_size,
                              hipStream_t stream) {
    const float* x   = (const float*)d_in[0];
    const int*   ei  = (const int*)d_in[1];
    const float* W1  = (const float*)d_in[2];
    const float* b1  = (const float*)d_in[3];
    const float* W2  = (const float*)d_in[4];
    const float* b2  = (const float*)d_in[5];
    float*       out = (float*)d_out;

    const int* srcp = ei;             // edge_index[0, :]
    const int* dstp = ei + N_EDGES;   // edge_index[1, :]

    // Workspace layout (256B aligned)
    char* ws = (char*)d_ws;
    float* dinv = (float*)ws;
    size_t off = (((size_t)N_NODES * sizeof(float)) + 255) & ~(size_t)255;
    float* bufA = (float*)(ws + off);                 // N x 128
    float* bufB = bufA + (size_t)N_NODES * 128;       // N x 128

    const int tiles = N_NODES / 16;                   // 6250
    const int gemm_blocks = (tiles + 7) / 8;
    const int edge_blocks = (N_EDGES + 7) / 8;

    // --- degree -> dinv ---
    hipMemsetAsync(dinv, 0, (size_t)N_NODES * sizeof(float), stream);
    gcn_deg_count<<<(N_EDGES + 255) / 256, 256, 0, stream>>>(dstp, dinv, N_EDGES);
    gcn_deg_to_dinv<<<(N_NODES + 255) / 256, 256, 0, stream>>>(dinv, N_NODES);

    // --- layer 1: h1 = relu(Agg(x@W1) + b1) ---
    gcn_gemm_wmma<128><<<gemm_blocks, 256, 0, stream>>>(x, W1, bufA, N_NODES);
    hipMemsetAsync(bufB, 0, (size_t)N_NODES * 128 * sizeof(float), stream);
    gcn_edge_scatter<128><<<edge_blocks, 256, 0, stream>>>(srcp, dstp, dinv, bufA, bufB, N_EDGES);
    {
        size_t total = (size_t)N_NODES * 128;
        gcn_finalize<128, true><<<(unsigned)((total + 255) / 256), 256, 0, stream>>>(
            bufB, bufA, dinv, b1, bufB, N_NODES);  // in-place into bufB
    }

    // --- layer 2: out = Agg(h1@W2) + b2 ---
    float* h2   = bufA;                               // N x 64
    float* agg2 = bufA + (size_t)N_NODES * 64;        // N x 64
    gcn_gemm_wmma<64><<<gemm_blocks, 256, 0, stream>>>(bufB, W2, h2, N_NODES);
    hipMemsetAsync(agg2, 0, (size_t)N_NODES * 64 * sizeof(float), stream);
    gcn_edge_scatter<64><<<edge_blocks, 256, 0, stream>>>(srcp, dstp, dinv, h2, agg2, N_EDGES);
    {
        size_t total = (size_t)N_NODES * 64;
        gcn_finalize<64, false><<<(unsigned)((total + 255) / 256), 256, 0, stream>>>(
            agg2, h2, dinv, b2, out, N_NODES);
    }
}